// Label_Propagation_78194174591125
// MI455X (gfx1250) — compile-verified
//
#include <hip/hip_runtime.h>

#define NN 8192
#define CC 64
#define EPSF 1e-12f
#define KP 256
#define LDS_STRIDE 260   // KP + 4: keeps float4 alignment, rotates LDS banks by 4/row

typedef __attribute__((ext_vector_type(2))) float v2f;
typedef __attribute__((ext_vector_type(4))) float v4f;
typedef __attribute__((ext_vector_type(8))) float v8f;

// ---------------- Kernel 1: inverse L1 row norms ----------------
__global__ void lp_rowsum_kernel(const float* __restrict__ bi,
                                 const float* __restrict__ mask,
                                 float* __restrict__ inv) {
    __shared__ float red[256];
    const int row = blockIdx.x;
    const int tid = threadIdx.x;
    const v4f* b4 = (const v4f*)(bi + (size_t)row * NN);
    const v4f* m4 = (const v4f*)(mask + (size_t)row * NN);
    float s = 0.0f;
    for (int j = tid; j < NN / 4; j += 256) {
        v4f b = __builtin_nontemporal_load(&b4[j]);
        v4f m = __builtin_nontemporal_load(&m4[j]);
        s += fabsf(b.x * m.x) + fabsf(b.y * m.y) +
             fabsf(b.z * m.z) + fabsf(b.w * m.w);
    }
    red[tid] = s;
    __syncthreads();
    for (int off = 128; off > 0; off >>= 1) {
        if (tid < off) red[tid] += red[tid + off];
        __syncthreads();
    }
    if (tid == 0) inv[row] = 1.0f / fmaxf(red[0], EPSF);
}

// ---------------- Kernel 2: normalize+store T, fused WMMA GEMM ----------------
// Block = 256 threads (8 wave32), owns a 16-row M-tile.
__global__ void __launch_bounds__(256)
lp_fused_kernel(const float* __restrict__ bi,
                const float* __restrict__ mask,
                const float* __restrict__ labels,
                const float* __restrict__ inv,
                float* __restrict__ Tout,
                float* __restrict__ yhat) {
    __shared__ float sT[16 * LDS_STRIDE];   // normalized T panel (16 x 256)
    __shared__ float sC[8 * 1024];          // per-wave partial 16x64 accumulators
    __shared__ float sInv[16];

    const int tid  = threadIdx.x;
    const int wave = tid >> 5;
    const int lane = tid & 31;
    const int row0 = blockIdx.x * 16;

    if (tid < 16) sInv[tid] = inv[row0 + tid];
    __syncthreads();

    v8f c0 = {}, c1 = {}, c2 = {}, c3 = {};

    const int m     = lane & 15;            // A row (and B column) index
    const int khalf = (lane >> 4) << 1;     // 0 for lanes 0-15, 2 for lanes 16-31

    for (int kbase = 0; kbase < NN; kbase += KP) {
        // ---- Phase A: stream panel, normalize, store T, stage in LDS ----
        #pragma unroll
        for (int it = 0; it < 4; ++it) {
            int r   = it * 4 + (tid >> 6);
            int col = (tid & 63) << 2;
            size_t g = (size_t)(row0 + r) * NN + (size_t)(kbase + col);
            v4f b  = __builtin_nontemporal_load((const v4f*)(bi + g));
            v4f mk = __builtin_nontemporal_load((const v4f*)(mask + g));
            float iv = sInv[r];
            v4f v;
            v.x = b.x * mk.x * iv;
            v.y = b.y * mk.y * iv;
            v.z = b.z * mk.z * iv;
            v.w = b.w * mk.w * iv;
            __builtin_nontemporal_store(v, (v4f*)(Tout + g));
            *(v4f*)(&sT[r * LDS_STRIDE + col]) = v;
        }
        __syncthreads();

        // ---- Phase B: each wave multiplies its 32-wide K slice ----
        const int ksub = wave * 32;
        #pragma unroll
        for (int step = 0; step < 8; ++step) {
            int kk = ksub + step * 4;
            // A fragment (16x4 f32): lane holds row m, K pair {khalf, khalf+1}
            v2f a = *(const v2f*)(&sT[m * LDS_STRIDE + kk + khalf]);
            // B fragments (4x16 f32): VGPR0 = K{0|2}, VGPR1 = K{1|3}, N = lane&15
            int gk = kbase + kk + khalf;
            const float* lb = labels + (size_t)gk * CC + m;
            v2f b0v, b1v, b2v, b3v;
            b0v.x = lb[0];   b0v.y = lb[CC + 0];
            b1v.x = lb[16];  b1v.y = lb[CC + 16];
            b2v.x = lb[32];  b2v.y = lb[CC + 32];
            b3v.x = lb[48];  b3v.y = lb[CC + 48];
            c0 = __builtin_amdgcn_wmma_f32_16x16x4_f32(false, a, false, b0v, (short)0, c0, false, false);
            c1 = __builtin_amdgcn_wmma_f32_16x16x4_f32(false, a, false, b1v, (short)0, c1, false, false);
            c2 = __builtin_amdgcn_wmma_f32_16x16x4_f32(false, a, false, b2v, (short)0, c2, false, false);
            c3 = __builtin_amdgcn_wmma_f32_16x16x4_f32(false, a, false, b3v, (short)0, c3, false, false);
        }
        __syncthreads();
    }

    // ---- Cross-wave reduction of partial y_hat tiles ----
    #pragma unroll
    for (int r = 0; r < 8; ++r) {
        sC[wave * 1024 + 0 * 256 + r * 32 + lane] = c0[r];
        sC[wave * 1024 + 1 * 256 + r * 32 + lane] = c1[r];
        sC[wave * 1024 + 2 * 256 + r * 32 + lane] = c2[r];
        sC[wave * 1024 + 3 * 256 + r * 32 + lane] = c3[r];
    }
    __syncthreads();

    for (int e = tid; e < 1024; e += 256) {
        float s = 0.0f;
        #pragma unroll
        for (int w = 0; w < 8; ++w) s += sC[w * 1024 + e];
        // Decompose per C/D VGPR layout: VGPR r, lanes 0-15 -> M=r, lanes 16-31 -> M=r+8
        int ntile = e >> 8;
        int rem   = e & 255;
        int r     = rem >> 5;
        int l     = rem & 31;
        int mm    = r + ((l >> 4) << 3);
        int nn    = ntile * 16 + (l & 15);
        yhat[(size_t)(row0 + mm) * CC + nn] = s;
    }
}

extern "C" void kernel_launch(void* const* d_in, const int* in_sizes, int n_in,
                              void* d_out, int out_size, void* d_ws, size_t ws_size,
                              hipStream_t stream) {
    // setup_inputs order: dataset, features, labels, si_adj, bi_adj, adjacency_mask
    const float* labels = (const float*)d_in[2];
    const float* bi_adj = (const float*)d_in[4];
    const float* mask   = (const float*)d_in[5];

    float* Tout = (float*)d_out;                  // N*N floats
    float* yhat = Tout + (size_t)NN * NN;         // N*C floats
    float* inv  = (float*)d_ws;                   // N floats of scratch

    lp_rowsum_kernel<<<NN, 256, 0, stream>>>(bi_adj, mask, inv);
    lp_fused_kernel<<<NN / 16, 256, 0, stream>>>(bi_adj, mask, labels, inv, Tout, yhat);
}